// KPlaneDensityField_88381837017835
// MI455X (gfx1250) — compile-verified
//
#include <hip/hip_runtime.h>
#include <hip/hip_bf16.h>

#define RES   256
#define FDIM  8
#define HID   64
#define PLANE_ELEMS (RES * RES * FDIM)   // 524288 floats = 2MB per plane

typedef float v2f __attribute__((ext_vector_type(2)));
typedef float v8f __attribute__((ext_vector_type(8)));

// ---------------------------------------------------------------------------
// Kernel 1: collapse the linear MLP:  wc[f] = sum_h w2[h] * w1[h, f]
// ---------------------------------------------------------------------------
__global__ void kp_wc_kernel(const float* __restrict__ w1,
                             const float* __restrict__ w2,
                             float* __restrict__ wc) {
    int f = threadIdx.x;
    if (f < FDIM) {
        float s = 0.f;
#pragma unroll
        for (int h = 0; h < HID; ++h)
            s = fmaf(w2[h], w1[h * FDIM + f], s);
        wc[f] = s;
    }
}

// ---------------------------------------------------------------------------
// Kernel 2: transpose grids [8,256,256] -> [256,256,8] (channel innermost)
// so each bilerp corner becomes one contiguous 32B read.
// ---------------------------------------------------------------------------
__global__ __launch_bounds__(256) void kp_transpose_kernel(
    const float* __restrict__ g01, const float* __restrict__ g02,
    const float* __restrict__ g12,
    float* __restrict__ T01, float* __restrict__ T02, float* __restrict__ T12) {
    int idx   = blockIdx.x * 256 + threadIdx.x;      // 0 .. 3*65536-1
    int plane = idx >> 16;
    int yx    = idx & 0xFFFF;
    const float* g = (plane == 0) ? g01 : (plane == 1) ? g02 : g12;
    float*       T = (plane == 0) ? T01 : (plane == 1) ? T02 : T12;
    float v[8];
#pragma unroll
    for (int c = 0; c < 8; ++c) v[c] = g[c * (RES * RES) + yx];   // coalesced per c
    float4* o = reinterpret_cast<float4*>(T + yx * 8);
    o[0] = make_float4(v[0], v[1], v[2], v[3]);
    o[1] = make_float4(v[4], v[5], v[6], v[7]);
}

// ---------------------------------------------------------------------------
// Helpers for the main kernel
// ---------------------------------------------------------------------------
__device__ __forceinline__ void load8(const float* __restrict__ p, float* __restrict__ o) {
    float4 a = *reinterpret_cast<const float4*>(p);
    float4 b = *reinterpret_cast<const float4*>(p + 4);
    o[0] = a.x; o[1] = a.y; o[2] = a.z; o[3] = a.w;
    o[4] = b.x; o[5] = b.y; o[6] = b.z; o[7] = b.w;
}

// 8-channel bilinear interpolation from a channel-innermost [256,256,8] grid.
__device__ __forceinline__ void bilerp8(const float* __restrict__ T,
                                        float x, float y, float* __restrict__ o) {
    float xp  = fminf(fmaxf((x + 1.f) * 127.5f, 0.f), 255.f);
    float yp  = fminf(fmaxf((y + 1.f) * 127.5f, 0.f), 255.f);
    float x0f = floorf(xp), y0f = floorf(yp);
    int   x0  = (int)x0f,   y0  = (int)y0f;
    int   x1  = min(x0 + 1, RES - 1);
    int   y1  = min(y0 + 1, RES - 1);
    float wx  = xp - x0f,   wy  = yp - y0f;
    int   r0  = y0 << 8,    r1  = y1 << 8;

    float c00[8], c01[8], c10[8], c11[8];
    load8(T + ((r0 + x0) << 3), c00);   // each corner: 2x global_load_b128
    load8(T + ((r0 + x1) << 3), c01);
    load8(T + ((r1 + x0) << 3), c10);
    load8(T + ((r1 + x1) << 3), c11);
#pragma unroll
    for (int c = 0; c < 8; ++c) {
        float top = fmaf(wx, c01[c] - c00[c], c00[c]);
        float bot = fmaf(wx, c11[c] - c10[c], c10[c]);
        o[c]      = fmaf(wy, bot - top, top);
    }
}

// ---------------------------------------------------------------------------
// Kernel 3: fused tri-plane gather + WMMA dot + exp.
// One lane per point; each wave finishes 32 points with two 16x16 WMMA tiles.
// ---------------------------------------------------------------------------
__global__ __launch_bounds__(256) void kp_main_kernel(
    const float* __restrict__ pts,  const float* __restrict__ aabb,
    const float* __restrict__ T01,  const float* __restrict__ T02,
    const float* __restrict__ T12,  const float* __restrict__ wc,
    float* __restrict__ out) {
    const int gid = blockIdx.x * 256 + threadIdx.x;

    // --- normalize point into [-1,1]^3 ---
    float lox = aabb[0], loy = aabb[1], loz = aabb[2];
    float hix = aabb[3], hiy = aabb[4], hiz = aabb[5];
    float sxn = 2.f / (hix - lox), syn = 2.f / (hiy - loy), szn = 2.f / (hiz - loz);
    float px = fmaf(pts[3 * gid + 0] - lox, sxn, -1.f);
    float py = fmaf(pts[3 * gid + 1] - loy, syn, -1.f);
    float pz = fmaf(pts[3 * gid + 2] - loz, szn, -1.f);

    // --- tri-plane features ---
    float f01[8], f02[8], f12[8], feat[8];
    bilerp8(T01, px, py, f01);
    bilerp8(T02, px, pz, f02);
    bilerp8(T12, py, pz, f12);
#pragma unroll
    for (int c = 0; c < 8; ++c) feat[c] = f01[c] * f02[c] * f12[c];

    // --- collapsed-MLP weights ---
    float w[8];
    load8(wc, w);

    // --- WMMA epilogue: per wave, two 16-point tiles, K=8 via two
    //     accumulating V_WMMA_F32_16X16X4_F32 each.
    // A 16x4 f32 layout: VGPR0 = K0 (lanes 0-15) / K2 (lanes 16-31),
    //                    VGPR1 = K1 / K3.  B mirrors it; B columns all = wc,
    //                    so every column of D holds the 16 dot products.
    const int  lane = threadIdx.x & 31;
    const bool hih  = lane >= 16;
    const int  srcA = lane & 15;         // tile0 sources: lanes 0..15
    const int  srcB = srcA + 16;         // tile1 sources: lanes 16..31

    float tA[8], tB[8];
#pragma unroll
    for (int c = 0; c < 8; ++c) {
        tA[c] = __shfl(feat[c], srcA, 32);
        tB[c] = __shfl(feat[c], srcB, 32);
    }

    v2f a0, a1, e0, e1, b0, b1;
    a0.x = hih ? tA[2] : tA[0];  a0.y = hih ? tA[3] : tA[1];
    a1.x = hih ? tA[6] : tA[4];  a1.y = hih ? tA[7] : tA[5];
    e0.x = hih ? tB[2] : tB[0];  e0.y = hih ? tB[3] : tB[1];
    e1.x = hih ? tB[6] : tB[4];  e1.y = hih ? tB[7] : tB[5];
    b0.x = hih ? w[2]  : w[0];   b0.y = hih ? w[3]  : w[1];
    b1.x = hih ? w[6]  : w[4];   b1.y = hih ? w[7]  : w[5];

    v8f acc0 = {};
    acc0 = __builtin_amdgcn_wmma_f32_16x16x4_f32(false, a0, false, b0,
                                                 (short)0, acc0, false, false);
    acc0 = __builtin_amdgcn_wmma_f32_16x16x4_f32(false, a1, false, b1,
                                                 (short)0, acc0, false, false);
    v8f acc1 = {};
    acc1 = __builtin_amdgcn_wmma_f32_16x16x4_f32(false, e0, false, b0,
                                                 (short)0, acc1, false, false);
    acc1 = __builtin_amdgcn_wmma_f32_16x16x4_f32(false, e1, false, b1,
                                                 (short)0, acc1, false, false);

    // --- store: D VGPR i holds M=i (lanes 0-15) / M=i+8 (lanes 16-31).
    // Lanes 0 and 16 each own 8 contiguous outputs per tile.
    const int wbase = blockIdx.x * 256 + (threadIdx.x & ~31);
    if ((lane & 15) == 0) {
        const int off = hih ? 8 : 0;
        float4 s0 = make_float4(expf(acc0[0]), expf(acc0[1]), expf(acc0[2]), expf(acc0[3]));
        float4 s1 = make_float4(expf(acc0[4]), expf(acc0[5]), expf(acc0[6]), expf(acc0[7]));
        float4 s2 = make_float4(expf(acc1[0]), expf(acc1[1]), expf(acc1[2]), expf(acc1[3]));
        float4 s3 = make_float4(expf(acc1[4]), expf(acc1[5]), expf(acc1[6]), expf(acc1[7]));
        *reinterpret_cast<float4*>(out + wbase + off)          = s0;  // tile0: pts wbase+0..15
        *reinterpret_cast<float4*>(out + wbase + off + 4)      = s1;
        *reinterpret_cast<float4*>(out + wbase + 16 + off)     = s2;  // tile1: pts wbase+16..31
        *reinterpret_cast<float4*>(out + wbase + 16 + off + 4) = s3;
    }
}

// ---------------------------------------------------------------------------
// Launch
// ---------------------------------------------------------------------------
extern "C" void kernel_launch(void* const* d_in, const int* in_sizes, int n_in,
                              void* d_out, int out_size, void* d_ws, size_t ws_size,
                              hipStream_t stream) {
    const float* pts  = (const float*)d_in[0];
    const float* aabb = (const float*)d_in[1];
    const float* g01  = (const float*)d_in[2];
    const float* g02  = (const float*)d_in[3];
    const float* g12  = (const float*)d_in[4];
    const float* w1   = (const float*)d_in[5];
    const float* w2   = (const float*)d_in[6];
    float*       out  = (float*)d_out;

    // workspace layout: [wc: 8 floats, padded to 4KB][T01][T02][T12] (~6.3MB)
    float* wsf = (float*)d_ws;
    float* wc  = wsf;
    float* T01 = wsf + 1024;
    float* T02 = T01 + PLANE_ELEMS;
    float* T12 = T02 + PLANE_ELEMS;

    kp_wc_kernel<<<1, 32, 0, stream>>>(w1, w2, wc);
    kp_transpose_kernel<<<(3 * RES * RES) / 256, 256, 0, stream>>>(
        g01, g02, g12, T01, T02, T12);

    const int n = in_sizes[0] / 3;   // 8192*256 points, divisible by 256
    kp_main_kernel<<<n / 256, 256, 0, stream>>>(pts, aabb, T01, T02, T12, wc, out);
}